// GCN1_21320217657891
// MI455X (gfx1250) — compile-verified
//
#include <hip/hip_runtime.h>
#include <math.h>

typedef __attribute__((ext_vector_type(2))) float v2f;
typedef __attribute__((ext_vector_type(8))) float v8f;

#define FIN  96
#define FMID 64
#define FOUT 16

// ---------- degree / norm ----------
__global__ void k_deg_init(float* __restrict__ deg, int n) {
    int i = blockIdx.x * blockDim.x + threadIdx.x;
    if (i < n) deg[i] = 1.0f;  // self-loop
}

__global__ void k_deg_edges(const int* __restrict__ dst, float* __restrict__ deg, int e) {
    int i = blockIdx.x * blockDim.x + threadIdx.x;
    if (i < e) atomicAdd(&deg[dst[i]], 1.0f);
}

__global__ void k_dinv(float* __restrict__ deg, int n) {
    int i = blockIdx.x * blockDim.x + threadIdx.x;
    if (i < n) deg[i] = rsqrtf(deg[i]);  // deg >= 1 always
}

// ---------- GEMM1: h1 = x @ W1 ; a1 = dinv^2 * h1 (self-loop seed) ----------
// one wave per 16-row tile, N=64 -> 4 output tiles, K=96 in steps of 4
__global__ __launch_bounds__(32) void k_gemm1(const float* __restrict__ x,
                                              const float* __restrict__ W1,
                                              const float* __restrict__ dinv,
                                              float* __restrict__ h1,
                                              float* __restrict__ a1, int n) {
    const int lane = threadIdx.x;
    const int lh = lane & 15;   // M (A) or N (B/C) index within tile
    const int kh = lane >> 4;   // K-half selector
    const int row0 = blockIdx.x * 16;

    int arow = row0 + lh; if (arow >= n) arow = n - 1;
    const float* ap = x + (size_t)arow * FIN + 2 * kh;

    v8f acc[4] = {};
#pragma unroll 4
    for (int k = 0; k < FIN; k += 4) {
        v2f a;
        a.x = ap[k];
        a.y = ap[k + 1];
        const int kk = k + 2 * kh;
#pragma unroll
        for (int t = 0; t < 4; ++t) {
            const float* bp = W1 + (size_t)kk * FMID + t * 16 + lh;
            v2f b;
            b.x = bp[0];
            b.y = bp[FMID];
            acc[t] = __builtin_amdgcn_wmma_f32_16x16x4_f32(
                false, a, false, b, (short)0, acc[t], false, false);
        }
    }

    float dv2[8];
#pragma unroll
    for (int j = 0; j < 8; ++j) {
        int m = row0 + j + 8 * kh;
        float dv = (m < n) ? dinv[m] : 0.0f;
        dv2[j] = dv * dv;
    }
#pragma unroll
    for (int t = 0; t < 4; ++t) {
#pragma unroll
        for (int j = 0; j < 8; ++j) {
            int m = row0 + j + 8 * kh;
            if (m < n) {
                size_t o = (size_t)m * FMID + t * 16 + lh;
                float v = acc[t][j];
                h1[o] = v;
                a1[o] = dv2[j] * v;
            }
        }
    }
}

// ---------- edge scatter layer 1: a1[dst] += dinv[s]*dinv[d]*h1[src] ----------
// 16 threads per edge, float4 per thread (64 features)
__global__ void k_agg1(const int* __restrict__ src, const int* __restrict__ dst,
                       const float* __restrict__ dinv, const float* __restrict__ h1,
                       float* __restrict__ a1, int e) {
    int i = blockIdx.x * blockDim.x + threadIdx.x;
    int idx = i >> 4;
    if (idx >= e) return;
    int f = (i & 15) * 4;
    int s = src[idx], d = dst[idx];
    float norm = dinv[s] * dinv[d];
    float4 hv = *(const float4*)(h1 + (size_t)s * FMID + f);
    float* op = a1 + (size_t)d * FMID + f;
    atomicAdd(op + 0, norm * hv.x);
    atomicAdd(op + 1, norm * hv.y);
    atomicAdd(op + 2, norm * hv.z);
    atomicAdd(op + 3, norm * hv.w);
}

// ---------- GEMM2: h2 = relu(a1+b1) @ W2 ; a2 = dinv^2 * h2 ----------
__global__ __launch_bounds__(32) void k_gemm2(const float* __restrict__ a1,
                                              const float* __restrict__ b1,
                                              const float* __restrict__ W2,
                                              const float* __restrict__ dinv,
                                              float* __restrict__ h2,
                                              float* __restrict__ a2, int n) {
    const int lane = threadIdx.x;
    const int lh = lane & 15;
    const int kh = lane >> 4;
    const int row0 = blockIdx.x * 16;

    int arow = row0 + lh; if (arow >= n) arow = n - 1;
    const float* ap = a1 + (size_t)arow * FMID + 2 * kh;

    v8f acc = {};
#pragma unroll 4
    for (int k = 0; k < FMID; k += 4) {
        const int kk = k + 2 * kh;
        v2f a;
        a.x = fmaxf(ap[k] + b1[kk], 0.0f);       // fused bias + ReLU
        a.y = fmaxf(ap[k + 1] + b1[kk + 1], 0.0f);
        const float* bp = W2 + (size_t)kk * FOUT + lh;
        v2f b;
        b.x = bp[0];
        b.y = bp[FOUT];
        acc = __builtin_amdgcn_wmma_f32_16x16x4_f32(
            false, a, false, b, (short)0, acc, false, false);
    }

#pragma unroll
    for (int j = 0; j < 8; ++j) {
        int m = row0 + j + 8 * kh;
        if (m < n) {
            float dv = dinv[m];
            size_t o = (size_t)m * FOUT + lh;
            float v = acc[j];
            h2[o] = v;
            a2[o] = dv * dv * v;
        }
    }
}

// ---------- edge scatter layer 2: 4 threads per edge ----------
__global__ void k_agg2(const int* __restrict__ src, const int* __restrict__ dst,
                       const float* __restrict__ dinv, const float* __restrict__ h2,
                       float* __restrict__ a2, int e) {
    int i = blockIdx.x * blockDim.x + threadIdx.x;
    int idx = i >> 2;
    if (idx >= e) return;
    int f = (i & 3) * 4;
    int s = src[idx], d = dst[idx];
    float norm = dinv[s] * dinv[d];
    float4 hv = *(const float4*)(h2 + (size_t)s * FOUT + f);
    float* op = a2 + (size_t)d * FOUT + f;
    atomicAdd(op + 0, norm * hv.x);
    atomicAdd(op + 1, norm * hv.y);
    atomicAdd(op + 2, norm * hv.z);
    atomicAdd(op + 3, norm * hv.w);
}

// ---------- bias + log_softmax ----------
__global__ void k_lsm(const float* __restrict__ a2, const float* __restrict__ b2,
                      float* __restrict__ out, int n) {
    int i = blockIdx.x * blockDim.x + threadIdx.x;
    if (i >= n) return;
    float v[FOUT];
    float m = -3.4e38f;
#pragma unroll
    for (int f = 0; f < FOUT; ++f) {
        v[f] = a2[(size_t)i * FOUT + f] + b2[f];
        m = fmaxf(m, v[f]);
    }
    float s = 0.0f;
#pragma unroll
    for (int f = 0; f < FOUT; ++f) s += __expf(v[f] - m);
    float l = __logf(s);
#pragma unroll
    for (int f = 0; f < FOUT; ++f) out[(size_t)i * FOUT + f] = v[f] - m - l;
}

extern "C" void kernel_launch(void* const* d_in, const int* in_sizes, int n_in,
                              void* d_out, int out_size, void* d_ws, size_t ws_size,
                              hipStream_t stream) {
    const float* x  = (const float*)d_in[0];
    const int*   ei = (const int*)d_in[1];
    const float* W1 = (const float*)d_in[2];
    const float* b1 = (const float*)d_in[3];
    const float* W2 = (const float*)d_in[4];
    const float* b2 = (const float*)d_in[5];

    const int n = in_sizes[0] / FIN;
    const int e = in_sizes[1] / 2;
    const int* src = ei;       // edge_index[0]
    const int* dst = ei + e;   // edge_index[1]

    float* ws   = (float*)d_ws;
    float* dinv = ws;                                // n
    float* h1   = dinv + n;                          // n*64
    float* a1   = h1 + (size_t)n * FMID;             // n*64
    float* h2   = a1 + (size_t)n * FMID;             // n*16
    float* a2   = h2 + (size_t)n * FOUT;             // n*16
    float* out  = (float*)d_out;

    const int B = 256;
    const int rowTiles = (n + 15) / 16;

    k_deg_init<<<(n + B - 1) / B, B, 0, stream>>>(dinv, n);
    k_deg_edges<<<(e + B - 1) / B, B, 0, stream>>>(dst, dinv, e);
    k_dinv<<<(n + B - 1) / B, B, 0, stream>>>(dinv, n);

    k_gemm1<<<rowTiles, 32, 0, stream>>>(x, W1, dinv, h1, a1, n);

    long items1 = (long)e * 16;
    k_agg1<<<(int)((items1 + B - 1) / B), B, 0, stream>>>(src, dst, dinv, h1, a1, e);

    k_gemm2<<<rowTiles, 32, 0, stream>>>(a1, b1, W2, dinv, h2, a2, n);

    long items2 = (long)e * 4;
    k_agg2<<<(int)((items2 + B - 1) / B), B, 0, stream>>>(src, dst, dinv, h2, a2, e);

    k_lsm<<<(n + B - 1) / B, B, 0, stream>>>(a2, b2, out, n);
}